// AtomDistances_42941083025444
// MI455X (gfx1250) — compile-verified
//
#include <hip/hip_runtime.h>
#include <hip/hip_bf16.h>
#include <stdint.h>

typedef uint32_t u32;
typedef uint64_t u64;
typedef __attribute__((ext_vector_type(4))) u32   v4u;
typedef __attribute__((ext_vector_type(4))) int   v4i;
typedef __attribute__((ext_vector_type(8))) int   v8i;
typedef __attribute__((ext_vector_type(4))) float v4f;

#define A_MAX    2048
#define TILE_I   16
#define NTHREADS 256
#define EPSF     1e-8f

#if defined(__HIP_DEVICE_COMPILE__) && __has_builtin(__builtin_amdgcn_tensor_load_to_lds) && __has_builtin(__builtin_amdgcn_s_wait_tensorcnt)
#define HAVE_TDM 1
#else
#define HAVE_TDM 0
#endif

#if HAVE_TDM
// Issue a 1-D Tensor-Data-Mover copy of `ndwords` dwords from global memory
// into LDS at byte offset lds_byte_off. Descriptor layout per CDNA5 ISA §8
// (CDNA5 ISA). Must be called wave-uniformly (EXEC ignored).
__device__ __forceinline__ void tdm_load_to_lds_1d(u32 lds_byte_off,
                                                   const void* gptr,
                                                   u32 ndwords) {
  u64 ga = (u64)(uintptr_t)gptr;
  // ---- D# group 0 (4 dwords) ----
  v4u g0;
  g0[0] = 0x1u;                                   // count=1 (valid), user mode
  g0[1] = lds_byte_off;                           // lds_addr
  g0[2] = (u32)ga;                                // global_addr[31:0]
  g0[3] = ((u32)(ga >> 32) & 0x01FFFFFFu)         // global_addr[56:32]
          | 0x80000000u;                          // type = 2 ("image")
  // ---- D# group 1 (8 dwords) ----
  v8i g1;
  g1[0] = (int)(2u << 16);                        // wg_mask=0 (not in cluster), data_size=2 -> 4B
  g1[1] = (int)((ndwords & 0xFFFFu) << 16);       // tensor_dim0 low16 (bits 63:48)
  g1[2] = (int)(((ndwords >> 16) & 0xFFFFu)       // tensor_dim0 high16 (bits 79:64)
          | (1u << 16));                          // tensor_dim1 = 1 (bits 95:80)
  g1[3] = (int)((ndwords & 0xFFFFu) << 16);       // tile_dim0 (bits 127:112)
  g1[4] = 1;                                      // tile_dim1 = 1, tile_dim2 = 0 (unused)
  g1[5] = (int)ndwords;                           // tensor_dim0_stride low32
  g1[6] = 0;                                      // stride0 hi16 = 0, stride1 lo16 = 0
  g1[7] = 0;                                      // stride1 hi32 = 0
  // ---- D# groups 2/3 unused (zeros: tile_dim3/4 = 0) ----
  v4i g2 = {0, 0, 0, 0};
  v4i g3 = {0, 0, 0, 0};
#if __clang_major__ >= 23
  v8i gx = {0, 0, 0, 0, 0, 0, 0, 0};              // clang-23 therock header arity
  __builtin_amdgcn_tensor_load_to_lds(g0, g1, g2, g3, gx, 0);
#else
  __builtin_amdgcn_tensor_load_to_lds(g0, g1, g2, g3, 0);  // ROCm 7.2 / clang-22 arity
#endif
}
#endif

// Raw CDNA5 v_sqrt_f32 / v_rcp_f32 (~1 ulp each) — plenty for a distance kernel,
// and avoids the precise-div / sqrt-fixup VALU expansions in the hot loop.
__device__ __forceinline__ float fast_inv(float dist_plus_eps) {
  return __builtin_amdgcn_rcpf(dist_plus_eps);
}
__device__ __forceinline__ float fast_sqrt(float x) {
  return __builtin_amdgcn_sqrtf(x);
}

__global__ __launch_bounds__(NTHREADS)
void AtomDistances_kernel(const float* __restrict__ positions,
                          const int*   __restrict__ neighbors,
                          const int*   __restrict__ neighbor_mask,
                          float*       __restrict__ out,
                          int A, int rows_per_blk) {
  __shared__ float s_pos[3 * A_MAX];   // AoS xyz, 24 KB @ A=2048
  __shared__ float s_maskf[A_MAX];     // mask as 0.0/1.0 multiplier, 8 KB

  const int b   = blockIdx.y;                 // batch from grid.y: no integer div
  const int i0  = blockIdx.x * rows_per_blk;  // row tile from grid.x
  const int tid = threadIdx.x;

  const float* posb  = positions     + (size_t)b * A * 3;
  const int*   maskb = neighbor_mask + (size_t)b * A;
  const bool   use_lds = (A <= A_MAX);

  if (use_lds) {
#if HAVE_TDM
    if (tid < 32) {  // wave 0 issues the DMA; uniform branch, EXEC ignored by TDM
      tdm_load_to_lds_1d((u32)(uintptr_t)&s_pos[0], posb, (u32)(3 * A));
    }
#else
    for (int k = tid; k < 3 * A; k += NTHREADS) s_pos[k] = posb[k];
#endif
    // Other waves convert the mask while the TDM runs.
    for (int j = tid; j < A; j += NTHREADS)
      s_maskf[j] = maskb[j] ? 1.0f : 0.0f;
#if HAVE_TDM
    if (tid < 32) __builtin_amdgcn_s_wait_tensorcnt(0);  // DMA landed in LDS
#endif
    __syncthreads();
  }

  const int vecs = A >> 2;  // float4 groups per row

  for (int ii = 0; ii < rows_per_blk; ++ii) {
    const int i = i0 + ii;
    if (i >= A) break;

    float cx, cy, cz, cm;
    if (use_lds) {
      cx = s_pos[3 * i + 0]; cy = s_pos[3 * i + 1]; cz = s_pos[3 * i + 2];
      cm = s_maskf[i];
    } else {
      cx = posb[3 * i + 0]; cy = posb[3 * i + 1]; cz = posb[3 * i + 2];
      cm = maskb[i] ? 1.0f : 0.0f;
    }

    const int* nrow = neighbors + ((size_t)b * A + i) * A;
    float*     orow = out       + ((size_t)b * A + i) * A;

    if (use_lds && ((A & 3) == 0)) {
      for (int v = tid; v < vecs; v += NTHREADS) {
        const int jb = v << 2;
        v4i n4 = __builtin_nontemporal_load((const v4i*)nrow + v);  // streaming TH=NT
        v4f m4 = *(const v4f*)(s_maskf + jb);                       // ds_load_b128
        v4f r;
#pragma unroll
        for (int k = 0; k < 4; ++k) {
          const int n = n4[k];
          const float dx = s_pos[3 * n + 0] - cx;
          const float dy = s_pos[3 * n + 1] - cy;
          const float dz = s_pos[3 * n + 2] - cz;
          const float d2 = dx * dx + dy * dy + dz * dz;
          float val = fast_inv(fast_sqrt(d2) + EPSF);  // d2==0 -> 1e8, matches ref
          val *= cm * m4[k];
          if (jb + k == i) val = 0.0f;                 // eye(A) diagonal
          r[k] = val;
        }
        __builtin_nontemporal_store(r, (v4f*)orow + v);             // streaming TH=NT
      }
    } else {
      for (int j = tid; j < A; j += NTHREADS) {
        const int n = nrow[j];
        float px, py, pz, mj;
        if (use_lds) {
          px = s_pos[3 * n + 0]; py = s_pos[3 * n + 1]; pz = s_pos[3 * n + 2];
          mj = s_maskf[j];
        } else {
          px = posb[3 * n + 0]; py = posb[3 * n + 1]; pz = posb[3 * n + 2];
          mj = maskb[j] ? 1.0f : 0.0f;
        }
        const float dx = px - cx, dy = py - cy, dz = pz - cz;
        const float d2 = dx * dx + dy * dy + dz * dz;
        float val = fast_inv(fast_sqrt(d2) + EPSF);
        val *= cm * mj;
        if (j == i) val = 0.0f;
        orow[j] = val;
      }
    }
  }
}

extern "C" void kernel_launch(void* const* d_in, const int* in_sizes, int n_in,
                              void* d_out, int out_size, void* d_ws, size_t ws_size,
                              hipStream_t stream) {
  const float* positions = (const float*)d_in[0];                 // (B, A, 3) f32
  const int*   neighbors = (const int*)d_in[1];                   // (B, A, A) int
  const int*   nmask     = (const int*)d_in[2];                   // (B, A)   int/bool
  float*       out       = (float*)d_out;                        // (B, A, A) f32

  // in_sizes: [B*A*3, B*A*A, B*A]  ->  A = (B*A*A)/(B*A), B = (B*A)/A
  const int A = in_sizes[1] / in_sizes[2];
  const int B = in_sizes[2] / A;

  const int rows_per_blk     = TILE_I;
  const int blocks_per_batch = (A + rows_per_blk - 1) / rows_per_blk;

  dim3 grid(blocks_per_batch, B);   // 2-D grid: batch = blockIdx.y (no int div on GPU)
  dim3 block(NTHREADS);
  hipLaunchKernelGGL(AtomDistances_kernel, grid, block, 0, stream,
                     positions, neighbors, nmask, out,
                     A, rows_per_blk);
}